// GraphAttention_20023137534775
// MI455X (gfx1250) — compile-verified
//
#include <hip/hip_runtime.h>
#include <math.h>

typedef float v2f __attribute__((ext_vector_type(2)));
typedef float v8f __attribute__((ext_vector_type(8)));

#define B_SZ 2048
#define N_SZ 8192
#define F_SZ 512
#define H_SZ 8
#define E_SZ 64
#define HP   16   // heads padded to 16 for WMMA

// ws layout in floats:
//  [0,      8192)   w2T[512][16]   (zero-padded cols h>=8, written by k_w2)
//  [8192,  73728)   e2[8][8192]
//  [73728, 204800)  pT[8192][16]   (zeroed by memset; softmax writes h<8)
//  [204800,212992)  g[16][512]     (zeroed by memset; atomically accumulated)
//  [212992,213504)  vr[512]
#define W2T_OFF 0
#define E2_OFF  8192
#define PT_OFF  73728
#define GACC_OFF 204800
#define VR_OFF  212992

// ---- w2T[k][h] = sum_e W[h][k][e] * att[h][64+e]  (zero for h>=8) ----
__global__ void k_w2(const float* __restrict__ W, const float* __restrict__ att,
                     float* __restrict__ w2T) {
    int t = blockIdx.x * blockDim.x + threadIdx.x;   // 0..8191
    int k = t >> 4;
    int h = t & 15;
    float acc = 0.f;
    if (h < H_SZ) {
        const float* wp = W + ((size_t)h * F_SZ + k) * E_SZ;
        const float* ap = att + h * 2 * E_SZ + E_SZ;
        #pragma unroll 8
        for (int e = 0; e < E_SZ; ++e) acc += wp[e] * ap[e];
    }
    w2T[k * HP + h] = acc;
}

// ---- e2[h][n] = sum_f G[n][f] * w2T[f][h]  via V_WMMA_F32_16X16X4_F32 ----
// One wave owns a 16-row tile of G; K loop over F=512 in steps of 4.
__global__ void k_e2(const float* __restrict__ G, const float* __restrict__ w2T,
                     float* __restrict__ e2) {
    int wave = threadIdx.x >> 5;
    int lane = threadIdx.x & 31;
    int c    = lane & 15;
    int koff = (lane >> 4) << 1;            // 0 for lanes 0-15, 2 for 16-31
    int rowbase = (blockIdx.x * 4 + wave) * 16;
    const float* Grow = G + (size_t)(rowbase + c) * F_SZ;
    v8f acc = {};
    for (int kk = 0; kk < F_SZ; kk += 4) {
        v2f a = *(const v2f*)(Grow + kk + koff);          // A: (M=c, K=kk+koff, kk+koff+1)
        v2f b;
        b.x = w2T[(kk + koff)     * HP + c];              // B: (K, N=c)
        b.y = w2T[(kk + koff + 1) * HP + c];
        acc = __builtin_amdgcn_wmma_f32_16x16x4_f32(
                  false, a, false, b, (short)0, acc, false, false);
    }
    if (c < H_SZ) {                                       // D: M=v (+8 for hi lanes), N=c
        int row0 = rowbase + ((lane >> 4) << 3);
        #pragma unroll
        for (int v = 0; v < 8; ++v)
            e2[(size_t)c * N_SZ + row0 + v] = acc[v];
    }
}

// ---- per-head softmax over n; write pT[n][h] (cols 8..15 stay zero) ----
__global__ void k_softmax(const float* __restrict__ e2, float* __restrict__ pT) {
    __shared__ float red[256];
    int h = blockIdx.x;
    int t = threadIdx.x;
    const float* row = e2 + (size_t)h * N_SZ;
    float m = -INFINITY;
    for (int i = t; i < N_SZ; i += 256) m = fmaxf(m, row[i]);
    red[t] = m; __syncthreads();
    for (int s = 128; s > 0; s >>= 1) { if (t < s) red[t] = fmaxf(red[t], red[t + s]); __syncthreads(); }
    m = red[0]; __syncthreads();
    float sum = 0.f;
    for (int i = t; i < N_SZ; i += 256) sum += expf(row[i] - m);
    red[t] = sum; __syncthreads();
    for (int s = 128; s > 0; s >>= 1) { if (t < s) red[t] += red[t + s]; __syncthreads(); }
    float inv = 1.0f / red[0];
    for (int i = t; i < N_SZ; i += 256) pT[i * HP + h] = expf(row[i] - m) * inv;
}

// ---- g[h][f] += sum_{n in chunk} pT[n][h] * G[n][f]  (WMMA, K-split + atomics) ----
__global__ void k_gacc(const float* __restrict__ G, const float* __restrict__ pT,
                       float* __restrict__ g) {
    int lane  = threadIdx.x & 31;
    int c     = lane & 15;
    int koff  = (lane >> 4) << 1;
    int fbase = blockIdx.x * 16;
    int k0    = blockIdx.y * 512;
    v8f acc = {};
    for (int kk = k0; kk < k0 + 512; kk += 4) {
        v2f a, b;
        a.x = pT[(kk + koff)     * HP + c];               // A: (M=head c, K)
        a.y = pT[(kk + koff + 1) * HP + c];
        b.x = G[(size_t)(kk + koff)     * F_SZ + fbase + c];  // B: (K, N=fbase+c)
        b.y = G[(size_t)(kk + koff + 1) * F_SZ + fbase + c];
        acc = __builtin_amdgcn_wmma_f32_16x16x4_f32(
                  false, a, false, b, (short)0, acc, false, false);
    }
    if (lane < 16) {                                      // rows 0..7 = real heads
        #pragma unroll
        for (int v = 0; v < 8; ++v)
            atomicAdd(&g[v * F_SZ + fbase + c], acc[v]);
    }
}

// ---- vr[h*64+e] = relu( sum_f g[h][f] * W[h][f][e] ) ----
__global__ void k_v(const float* __restrict__ g, const float* __restrict__ W,
                    float* __restrict__ vr) {
    int t = threadIdx.x;           // 512 threads
    int h = t >> 6, e = t & 63;
    const float* gr = g + h * F_SZ;
    const float* wp = W + (size_t)h * F_SZ * E_SZ + e;
    float acc = 0.f;
    for (int f = 0; f < F_SZ; ++f) acc += gr[f] * wp[(size_t)f * E_SZ];
    vr[t] = fmaxf(acc, 0.f);
}

// ---- out[b][:] = vr  (broadcast, 4MB, float4 stores) ----
__global__ void k_bcast(const float4* __restrict__ vr4, float4* __restrict__ out4) {
    int idx = blockIdx.x * blockDim.x + threadIdx.x;   // 262144 total
    out4[idx] = vr4[idx & 127];
}

extern "C" void kernel_launch(void* const* d_in, const int* in_sizes, int n_in,
                              void* d_out, int out_size, void* d_ws, size_t ws_size,
                              hipStream_t stream) {
    (void)in_sizes; (void)n_in; (void)out_size; (void)ws_size;
    // d_in[0] = X (provably unused: softmax over n cancels the e1[b] term)
    const float* G   = (const float*)d_in[1];
    const float* W   = (const float*)d_in[2];
    const float* att = (const float*)d_in[3];
    float* out = (float*)d_out;
    float* ws  = (float*)d_ws;

    float* w2T = ws + W2T_OFF;
    float* e2  = ws + E2_OFF;
    float* pT  = ws + PT_OFF;
    float* g   = ws + GACC_OFF;
    float* vr  = ws + VR_OFF;

    // zero pT (padding cols) and g (atomic accumulator) — contiguous region
    hipMemsetAsync(pT, 0, (size_t)(131072 + 8192) * sizeof(float), stream);

    k_w2     <<<32, 256, 0, stream>>>(W, att, w2T);
    k_e2     <<<128, 128, 0, stream>>>(G, w2T, e2);            // 512 waves, 16 rows each
    k_softmax<<<H_SZ, 256, 0, stream>>>(e2, pT);
    k_gacc   <<<dim3(32, 16), 32, 0, stream>>>(G, pT, g);      // 32 f-tiles x 16 K-chunks
    k_v      <<<1, 512, 0, stream>>>(g, W, vr);
    k_bcast  <<<1024, 256, 0, stream>>>((const float4*)vr, (float4*)out);
}